// CausalConvolution_56581899157889
// MI455X (gfx1250) — compile-verified
//
#include <hip/hip_runtime.h>
#include <hip/hip_bf16.h>

typedef __attribute__((ext_vector_type(2))) float v2f;
typedef __attribute__((ext_vector_type(8))) float v8f;

#define BB 16
#define HH 4
#define SS 8
#define WW 256
#define FF 16

// One block per (b, h, x, y). 256 threads = 8 wave32s.
// Wave w computes output M-tiles m=w and m=15-w (rows j in [16m,16m+16)).
// Y[j,f] = sum_{i<=j} k[(i-j-1)&255] * X[i,f]  via v_wmma_f32_16x16x4_f32,
// generating the triangular-Toeplitz A fragments on the fly from LDS.
__global__ __launch_bounds__(256) void causal_conv_wmma(
    const float* __restrict__ x,     // [B,S,W,F]
    const float* __restrict__ kern,  // [H,S,S,W]
    float* __restrict__ out)         // [B,H,S,S,W,F]
{
  __shared__ float sK[WW];                         // kernel slice, 1 KB
  __shared__ __attribute__((aligned(16))) float sX[WW * FF];  // x slice, 16 KB

  const int blk = blockIdx.x;        // 0..4095
  const int b   = blk >> 8;
  const int hxy = blk & 255;
  const int h   = hxy >> 6;
  const int sx  = (hxy >> 3) & 7;
  const int sy  = hxy & 7;

  const int tid = threadIdx.x;       // 0..255

  // Stage kernel slice k[h,sx,sy,:] (256 floats)
  sK[tid] = kern[((h * SS + sx) * SS + sy) * WW + tid];

  // Stage X slice x[b,sx,:,:] (256x16 floats) with float4 copies
  {
    const float4* xsrc = (const float4*)(x + ((size_t)b * SS + sx) * (WW * FF));
    float4* xdst = (float4*)sX;
#pragma unroll
    for (int t = 0; t < 4; ++t) xdst[tid + 256 * t] = xsrc[tid + 256 * t];
  }
  __syncthreads();

  const int wave  = tid >> 5;        // 0..7 (uniform per wave)
  const int lane  = tid & 31;
  const int lhalf = lane >> 4;       // 0|1
  const int ln    = lane & 15;

  const int mA = wave;               // 0..7
  const int mB = 15 - wave;          // 15..8
  const int jA = 16 * mA + ln;       // A-fragment row for this lane (M = ln)
  const int jB = 16 * mB + ln;

  v8f accA = {};
  v8f accB = {};

  // K-tiles shared by both M-tiles (kt <= mA < mB): B fragment reused,
  // triangular mask only matters on kt==mA for accA.
  for (int kt = 0; kt <= mA; ++kt) {
#pragma unroll
    for (int s = 0; s < 4; ++s) {
      const int i0 = 16 * kt + 4 * s + 2 * lhalf;   // K = {i0, i0+1} per lane
      v2f bm;
      bm.x = sX[i0 * FF + ln];
      bm.y = sX[(i0 + 1) * FF + ln];
      v2f aA;
      aA.x = (i0     <= jA) ? sK[(i0 - jA - 1) & 255] : 0.0f;
      aA.y = (i0 + 1 <= jA) ? sK[(i0 - jA)     & 255] : 0.0f;
      accA = __builtin_amdgcn_wmma_f32_16x16x4_f32(
          false, aA, false, bm, (short)0, accA, false, false);
      v2f aB;                         // kt <= mA < mB  =>  i always <= jB
      aB.x = sK[(i0 - jB - 1) & 255];
      aB.y = sK[(i0 - jB)     & 255];
      accB = __builtin_amdgcn_wmma_f32_16x16x4_f32(
          false, aB, false, bm, (short)0, accB, false, false);
    }
  }
  // Remaining K-tiles for the far M-tile (mask matters only at kt==mB).
  for (int kt = mA + 1; kt <= mB; ++kt) {
#pragma unroll
    for (int s = 0; s < 4; ++s) {
      const int i0 = 16 * kt + 4 * s + 2 * lhalf;
      v2f bm;
      bm.x = sX[i0 * FF + ln];
      bm.y = sX[(i0 + 1) * FF + ln];
      v2f aB;
      aB.x = (i0     <= jB) ? sK[(i0 - jB - 1) & 255] : 0.0f;
      aB.y = (i0 + 1 <= jB) ? sK[(i0 - jB)     & 255] : 0.0f;
      accB = __builtin_amdgcn_wmma_f32_16x16x4_f32(
          false, aB, false, bm, (short)0, accB, false, false);
    }
  }

  // Epilogue: divide by (j+1); on the diagonal series roll time by 1 and zero j=0.
  float* outp = out + ((((size_t)b * HH + h) * SS + sx) * SS + sy) * (size_t)(WW * FF);
  const bool diag = (sx == sy);

#pragma unroll
  for (int v = 0; v < 8; ++v) {
    const int jj = 16 * mA + v + 8 * lhalf;       // C/D layout: M = v + 8*half
    const float val = accA[v] / (float)(jj + 1);
    if (!diag) {
      outp[jj * FF + ln] = val;
    } else {
      if (jj + 1 < WW) outp[(jj + 1) * FF + ln] = val;   // rolled: row j+1 = Y[j]/(j+1)
      if (jj == 0)     outp[ln] = 0.0f;                  // first step zeroed
    }
  }
#pragma unroll
  for (int v = 0; v < 8; ++v) {
    const int jj = 16 * mB + v + 8 * lhalf;
    const float val = accB[v] / (float)(jj + 1);
    if (!diag) {
      outp[jj * FF + ln] = val;
    } else {
      if (jj + 1 < WW) outp[(jj + 1) * FF + ln] = val;
    }
  }
}

extern "C" void kernel_launch(void* const* d_in, const int* in_sizes, int n_in,
                              void* d_out, int out_size, void* d_ws, size_t ws_size,
                              hipStream_t stream) {
  const float* x    = (const float*)d_in[0];   // [16,8,256,16]
  const float* kern = (const float*)d_in[1];   // [4,8,8,256]
  float* out        = (float*)d_out;           // [16,4,8,8,256,16]
  (void)in_sizes; (void)n_in; (void)out_size; (void)d_ws; (void)ws_size;

  dim3 grid(BB * HH * SS * SS);   // 4096 blocks: (b,h,x,y)
  dim3 block(256);                // 8 wave32s
  causal_conv_wmma<<<grid, block, 0, stream>>>(x, kern, out);
}